// GhostVLAD_82575041233229
// MI455X (gfx1250) — compile-verified
//
#include <hip/hip_runtime.h>
#include <hip/hip_bf16.h>

typedef __attribute__((ext_vector_type(2))) float v2f;
typedef __attribute__((ext_vector_type(8))) float v8f;

#define NBATCH   32
#define CDIM     512
#define LPOS     512
#define KG       10
#define KP       16
#define VLADK    8
#define EPSF     1e-12f
#define NTHREADS 512
#define NWAVES   (NTHREADS / 32)

__global__ __launch_bounds__(NTHREADS)
void ghostvlad_kernel(const float* __restrict__ x,
                      const float* __restrict__ conv_w,
                      const float* __restrict__ conv_b,
                      const float* __restrict__ centers,
                      float* __restrict__ out)
{
    __shared__ float sA[KP * LPOS];      // logits, then softmax assignments a  (32 KB)
    __shared__ float sRes[VLADK * CDIM]; // residuals for the 8 real clusters   (16 KB)
    __shared__ float sBias[KP];
    __shared__ float sAsum[KP];
    __shared__ float sSq[VLADK];

    const int tid  = threadIdx.x;
    const int wave = tid >> 5;
    const int lane = tid & 31;
    const int p    = lane >> 4;   // half-wave select (fragment K / M offset)
    const int r    = lane & 15;   // row/col within fragment

    const int n = blockIdx.x;
    const float* __restrict__ xn = x + (size_t)n * CDIM * LPOS;

    if (tid < KP) {
        sBias[tid] = (tid < KG) ? conv_b[tid] : 0.0f;
        sAsum[tid] = 0.0f;
    }
    if (tid < VLADK) sSq[tid] = 0.0f;
    __syncthreads();

    // Branchless padding for conv_w rows 10..15: clamp row, scale by 0.
    const int   rw     = (r < KG) ? r : 0;
    const float wscale = (r < KG) ? 1.0f : 0.0f;

    // ---------------- Phase A: logits[16 x 512] = conv_w_pad * x ----------------
    // A = conv_w (M=cluster, K=c), B = x (K=c, N=l).  K stepped by 4 (f32 WMMA).
    for (int t = wave; t < LPOS / 16; t += NWAVES) {
        const int l0 = t * 16;
        v8f acc = {};
        for (int kc = 0; kc < CDIM / 4; ++kc) {
            const int cb = kc * 4 + 2 * p;
            v2f av, bv;
            av.x = conv_w[rw * CDIM + cb]     * wscale;
            av.y = conv_w[rw * CDIM + cb + 1] * wscale;
            bv.x = xn[(size_t)(cb)     * LPOS + l0 + r];
            bv.y = xn[(size_t)(cb + 1) * LPOS + l0 + r];
            acc = __builtin_amdgcn_wmma_f32_16x16x4_f32(
                false, av, false, bv, (short)0, acc, false, false);
        }
        #pragma unroll
        for (int j = 0; j < 8; ++j) {
            const int k = j + 8 * p;                 // D row = M
            sA[k * LPOS + l0 + r] = acc[j] + sBias[k];
        }
    }
    __syncthreads();

    // ---------------- Softmax over k (10 valid), zero pad rows ----------------
    {
        const int l = tid;                           // 512 threads, 512 positions
        float m = -3.402823e38f;
        #pragma unroll
        for (int k = 0; k < KG; ++k) m = fmaxf(m, sA[k * LPOS + l]);
        float e[KG];
        float s = 0.0f;
        #pragma unroll
        for (int k = 0; k < KG; ++k) { e[k] = __expf(sA[k * LPOS + l] - m); s += e[k]; }
        const float inv = 1.0f / s;
        #pragma unroll
        for (int k = 0; k < KG; ++k) sA[k * LPOS + l] = e[k] * inv;
        #pragma unroll
        for (int k = KG; k < KP; ++k) sA[k * LPOS + l] = 0.0f;
    }
    __syncthreads();

    // ---------------- assign_sum[k] = sum_l a[k][l] ----------------
    {
        const int k   = tid & 15;
        const int seg = tid >> 4;                    // 0..31, 16 l each
        float part = 0.0f;
        for (int l = seg * 16; l < seg * 16 + 16; ++l) part += sA[k * LPOS + l];
        atomicAdd(&sAsum[k], part);
    }
    __syncthreads();

    // ---------------- Phase B: assign[16 x 512] = a * x^T, residual ----------------
    // A = a (M=cluster, K=l), B = x^T (K=l, N=c).  B pairs are contiguous in memory.
    for (int t = wave; t < CDIM / 16; t += NWAVES) {
        const int c0 = t * 16;
        v8f acc = {};
        for (int kc = 0; kc < LPOS / 4; ++kc) {
            const int lb = kc * 4 + 2 * p;
            v2f av, bv;
            av.x = sA[r * LPOS + lb];
            av.y = sA[r * LPOS + lb + 1];
            const float* bp = xn + (size_t)(c0 + r) * LPOS + lb;
            bv.x = bp[0];
            bv.y = bp[1];
            acc = __builtin_amdgcn_wmma_f32_16x16x4_f32(
                false, av, false, bv, (short)0, acc, false, false);
        }
        if (p == 0) {                                // M=0..7: the 8 real clusters
            #pragma unroll
            for (int j = 0; j < 8; ++j) {
                const int c = c0 + r;
                sRes[j * CDIM + c] = acc[j] - sAsum[j] * centers[j * CDIM + c];
            }
        }
    }
    __syncthreads();

    // ---------------- per-cluster squared norms ----------------
    {
        const int k   = tid >> 6;                    // 0..7
        const int seg = tid & 63;                    // 64 segments of 8
        float part = 0.0f;
        for (int c = seg * 8; c < seg * 8 + 8; ++c) {
            const float v = sRes[k * CDIM + c];
            part += v * v;
        }
        atomicAdd(&sSq[k], part);
    }
    __syncthreads();

    // ---------------- intra-norm, global norm, write out ----------------
    float tot = 0.0f;
    #pragma unroll
    for (int k = 0; k < VLADK; ++k) {
        const float nk = fmaxf(sqrtf(sSq[k]), EPSF);
        tot += sSq[k] / (nk * nk);
    }
    const float tinv = 1.0f / fmaxf(sqrtf(tot), EPSF);

    float* __restrict__ on = out + (size_t)n * (VLADK * CDIM);
    #pragma unroll
    for (int i = 0; i < (VLADK * CDIM) / NTHREADS; ++i) {
        const int idx = i * NTHREADS + tid;          // coalesced
        const int k   = idx >> 9;
        const float nk = fmaxf(sqrtf(sSq[k]), EPSF);
        on[idx] = sRes[idx] * (1.0f / nk) * tinv;
    }
}

extern "C" void kernel_launch(void* const* d_in, const int* in_sizes, int n_in,
                              void* d_out, int out_size, void* d_ws, size_t ws_size,
                              hipStream_t stream) {
    const float* x      = (const float*)d_in[0];   // (32, 512, 16, 32)
    const float* conv_w = (const float*)d_in[1];   // (10, 512)
    const float* conv_b = (const float*)d_in[2];   // (10,)
    const float* cent   = (const float*)d_in[3];   // (10, 512)
    float* out = (float*)d_out;                    // (32, 4096)
    ghostvlad_kernel<<<NBATCH, NTHREADS, 0, stream>>>(x, conv_w, conv_b, cent, out);
}